// QuantumLayer_39341900431498
// MI455X (gfx1250) — compile-verified
//
#include <hip/hip_runtime.h>

// ---------------------------------------------------------------------------
// out[b][q] = cos(theta_q) * sum_k sign_q(k) x[b][k]^2 / sum_k x[b][k]^2
//   sign_q(k) = 1 - 2*((k >> (3-q)) & 1)        (wire 0 = MSB of state index)
//
// Implemented as D = S^T (16x16, constant) x Xsq^T (16x16, per 16-item tile)
// via a chain of four V_WMMA_F32_16X16X4_F32 ops (K = 16 split into 4x4).
// D layout (32-bit C/D 16x16): lane n (0..15) holds column n = item n,
// VGPR v holds row v -> acc[0..3] = signed sums, acc[4] = squared norm,
// all resident in ONE lane -> normalize + cos scale + float4 store, no LDS.
// ---------------------------------------------------------------------------

typedef __attribute__((ext_vector_type(2))) float v2f;
typedef __attribute__((ext_vector_type(8))) float v8f;

__device__ __forceinline__ float amat(int m, int k) {
  // S^T[m][k]: rows 0..3 = sign patterns for Z_q, row 4 = ones (norm), rest 0.
  if (m < 4) return ((k >> (3 - m)) & 1) ? -1.0f : 1.0f;
  if (m == 4) return 1.0f;
  return 0.0f;
}

__global__ __launch_bounds__(256) void qz16_wmma_kernel(
    const float* __restrict__ x,    // [BATCH, 16]
    const float* __restrict__ qw,   // [4]
    float* __restrict__ out,        // [BATCH, 4]
    int nTiles)                     // BATCH / 16
{
  const int lane   = threadIdx.x & 31;
  const int waveId = (int)((blockIdx.x * blockDim.x + threadIdx.x) >> 5);
  const int nWaves = (int)((gridDim.x * blockDim.x) >> 5);

  // cos(full angle) per wire — uniform, computed once per thread.
  const float c0 = __builtin_cosf(qw[0]);
  const float c1 = __builtin_cosf(qw[1]);
  const float c2 = __builtin_cosf(qw[2]);
  const float c3 = __builtin_cosf(qw[3]);

  // Constant A-matrix (S^T) chunks, 16x4 f32 layout:
  //   VGPR0: K=4kk+0 (lanes 0-15) / K=4kk+2 (lanes 16-31); VGPR1: +1.
  const int m     = lane & 15;         // A row (lanes 0-15 and 16-31 share M)
  const int khalf = (lane >> 4) << 1;  // 0 for lanes 0-15, 2 for lanes 16-31
  v2f A[4];
#pragma unroll
  for (int kk = 0; kk < 4; ++kk) {
    const int k0 = 4 * kk + khalf;
    A[kk].x = amat(m, k0);
    A[kk].y = amat(m, k0 + 1);
  }

  for (int t = waveId; t < nTiles; t += nWaves) {
    const float* base = x + (size_t)t * 256;  // 16 items * 16 floats

    // B-matrix = Xsq^T chunks (4x16, assumed mirror of A layout):
    //   VGPR0: K=4kk+0 / 4kk+2 ; VGPR1: K=4kk+1 / 4kk+3 ; N = item = lane&15.
    // Lane n grabs float2 at state offset (4kk + khalf) of item n: the four
    // b64 loads of a wave jointly consume every byte of the 1 KB tile.
    v2f B[4];
#pragma unroll
    for (int kk = 0; kk < 4; ++kk) {
      const float2 d = *(const float2*)(base + m * 16 + 4 * kk + khalf);
      B[kk].x = d.x * d.x;
      B[kk].y = d.y * d.y;
    }

    v8f acc = {};
#pragma unroll
    for (int kk = 0; kk < 4; ++kk) {
      // D = A x B + C ; 8 args: (neg_a, A, neg_b, B, c_mod, C, reuse_a, reuse_b)
      acc = __builtin_amdgcn_wmma_f32_16x16x4_f32(
          false, A[kk], false, B[kk], (short)0, acc, false, false);
    }

    if (lane < 16) {  // lanes 16-31 hold D rows 8-15 (all zero)
      const float r = __builtin_amdgcn_rcpf(acc[4]);  // 1 / ||x||^2
      float4 o;
      o.x = c0 * acc[0] * r;
      o.y = c1 * acc[1] * r;
      o.z = c2 * acc[2] * r;
      o.w = c3 * acc[3] * r;
      *(float4*)(out + (size_t)t * 64 + (size_t)lane * 4) = o;  // coalesced b128
    }
  }
}

extern "C" void kernel_launch(void* const* d_in, const int* in_sizes, int n_in,
                              void* d_out, int out_size, void* d_ws, size_t ws_size,
                              hipStream_t stream) {
  const float* x  = (const float*)d_in[0];   // [BATCH,16] f32
  const float* qw = (const float*)d_in[1];   // [4] f32
  float* out      = (float*)d_out;           // [BATCH,4] f32

  const int nTiles = in_sizes[0] / 256;      // (BATCH*16)/(16 items * 16 elems)

  // Persistent waves: 2048 blocks x 256 thr = 16384 waves, ~16 tiles each.
  dim3 grid(2048), block(256);
  hipLaunchKernelGGL(qz16_wmma_kernel, grid, block, 0, stream, x, qw, out, nTiles);
}